// Slicing_45217415692505
// MI455X (gfx1250) — compile-verified
//
#include <hip/hip_runtime.h>

// Bilateral-grid slicing: grid (16,12,8,16,16) f32, guide (16,1,512,512) f32
// -> out (16,12,512,512) f32.
// Memory-bound: ~220MB @ 23.3 TB/s ~= 9.4us floor. The LDS gather is the
// secondary limiter, so the grid is re-laid-out as [y][x][z][c] (z & channel
// in the low address bits -> per-lane-random banks, no ix-driven 8-way
// conflicts) and read as ds_load_b128 channel quads (24 loads/px vs 96).
// CDNA5 paths: async global->LDS DMA (ASYNCcnt) of the 96KB per-batch grid,
// wave32 float4-coalesced nontemporal streaming I/O.

#define BLOCKT   256
#define GH       512
#define GW       512
#define GD       8
#define GRH      16
#define GRW      16
#define CH       12
#define CELLS    (GD * GRH * GRW)          // 2048 spatial cells
#define GRID_PER_B (CH * CELLS)            // 24576 floats = 96 KB

typedef float v4f __attribute__((ext_vector_type(4)));

// ---- one-time transpose: grid[b][c][z][y][x] -> ws[b][y][x][z][c] ---------
__global__ __launch_bounds__(BLOCKT) void transpose_kernel(
    const float* __restrict__ grid, float* __restrict__ ws) {
  int tid = blockIdx.x * BLOCKT + threadIdx.x;     // 16*24576 total
  int b = tid / GRID_PER_B;
  int r = tid - b * GRID_PER_B;
  int c = r / CELLS;
  int p = r - c * CELLS;
  int z = p >> 8;
  int y = (p >> 4) & 15;
  int x = p & 15;
  ws[(size_t)b * GRID_PER_B + (size_t)(((y * GRW + x) * GD + z) * CH + c)] =
      grid[tid];
}

__global__ __launch_bounds__(BLOCKT) void slice_kernel(
    const float* __restrict__ grid,
    const float* __restrict__ guide,
    const float* __restrict__ wsT,     // pre-transposed grid (may be null)
    float* __restrict__ out,
    int use_ws) {
  __shared__ float sgT[GRID_PER_B];    // [y][x][z][c], rows of 12 floats

  const int b    = blockIdx.x >> 6;    // 64 tiles per batch
  const int tile = blockIdx.x & 63;    // tile = 8 consecutive rows
  const int t    = threadIdx.x;

  if (use_ws) {
    // ---- CDNA5 async global->LDS DMA of the pre-transposed 96 KB grid ----
    unsigned long long gbase =
        (unsigned long long)(const void*)(wsT + (size_t)b * GRID_PER_B);
    unsigned lds0 = (unsigned)(size_t)(&sgT[0]);
    #pragma unroll
    for (int i = 0; i < 24; ++i) {               // 24*256*16B = 98304 B
      unsigned byteOff = (unsigned)((i * BLOCKT + t) * 16);
      unsigned lds_addr = lds0 + byteOff;
      unsigned long long gaddr = gbase + (unsigned long long)byteOff;
      asm volatile("global_load_async_to_lds_b128 %0, %1, off"
                   :: "v"(lds_addr), "v"(gaddr) : "memory");
    }
    asm volatile("s_wait_asynccnt 0" ::: "memory");
  } else {
    // ---- fallback: transpose in-block (coalesced b128 reads, ds scatter) --
    const float* gsrc = grid + (size_t)b * GRID_PER_B;
    #pragma unroll
    for (int i = 0; i < 24; ++i) {
      int e4 = (i * BLOCKT + t) * 4;             // 4 consecutive x
      v4f a = *(const v4f*)(gsrc + e4);
      int c = e4 / CELLS;
      int p = e4 - c * CELLS;
      int z = p >> 8;
      int y = (p >> 4) & 15;
      int x = p & 15;
      #pragma unroll
      for (int j = 0; j < 4; ++j)
        sgT[((y * GRW + x + j) * GD + z) * CH + c] = a[j];
    }
  }
  __syncthreads();

  const float* gguide = guide + (size_t)b * (GH * GW);
  float*       gout   = out   + (size_t)b * CH * GH * GW;

  #pragma unroll 1
  for (int it = 0; it < 4; ++it) {
    const int g  = it * BLOCKT + t;        // group id within tile [0,1024)
    const int y  = tile * 8 + (g >> 7);    // 8 rows per tile
    const int x4 = (g & 127) * 4;          // 4 consecutive pixels / thread

    v4f gz = __builtin_nontemporal_load(
        (const v4f*)(gguide + (size_t)y * GW + x4));

    // y interpolation terms (shared by the 4 pixels of this thread)
    float yn   = ((float)y - 256.0f) * (1.0f / 256.0f);
    float iy   = ((yn + 1.0f) * 16.0f - 1.0f) * 0.5f;
    iy         = fminf(fmaxf(iy, 0.0f), 15.0f);
    float iy0f = floorf(iy);
    float fy   = iy - iy0f;
    int   iy0  = (int)iy0f;
    int   iy1  = min(iy0 + 1, 15);

    int   off[4][8];   // word offsets of the 8 corners ([y][x][z][c] layout)
    float w[4][8];
    #pragma unroll
    for (int j = 0; j < 4; ++j) {
      int   xi   = x4 + j;
      float xn   = ((float)xi - 256.0f) * (1.0f / 256.0f);
      float ix   = ((xn + 1.0f) * 16.0f - 1.0f) * 0.5f;
      ix         = fminf(fmaxf(ix, 0.0f), 15.0f);
      float ix0f = floorf(ix);
      float fx   = ix - ix0f;
      int   ix0  = (int)ix0f;
      int   ix1  = min(ix0 + 1, 15);

      float z    = gz[j];
      float iz   = ((z + 1.0f) * 8.0f - 1.0f) * 0.5f;
      iz         = fminf(fmaxf(iz, 0.0f), 7.0f);
      float iz0f = floorf(iz);
      float fz   = iz - iz0f;
      int   iz0  = (int)iz0f;
      int   iz1  = min(iz0 + 1, 7);

      float wx0 = 1.0f - fx, wx1 = fx;
      float wy0 = 1.0f - fy, wy1 = fy;
      float wz0 = 1.0f - fz, wz1 = fz;

      int oy0x0 = (iy0 * GRW + ix0) * GD;
      int oy0x1 = (iy0 * GRW + ix1) * GD;
      int oy1x0 = (iy1 * GRW + ix0) * GD;
      int oy1x1 = (iy1 * GRW + ix1) * GD;

      off[j][0] = (oy0x0 + iz0) * CH;  w[j][0] = wz0 * wy0 * wx0;
      off[j][1] = (oy0x0 + iz1) * CH;  w[j][1] = wz1 * wy0 * wx0;
      off[j][2] = (oy0x1 + iz0) * CH;  w[j][2] = wz0 * wy0 * wx1;
      off[j][3] = (oy0x1 + iz1) * CH;  w[j][3] = wz1 * wy0 * wx1;
      off[j][4] = (oy1x0 + iz0) * CH;  w[j][4] = wz0 * wy1 * wx0;
      off[j][5] = (oy1x0 + iz1) * CH;  w[j][5] = wz1 * wy1 * wx0;
      off[j][6] = (oy1x1 + iz0) * CH;  w[j][6] = wz0 * wy1 * wx1;
      off[j][7] = (oy1x1 + iz1) * CH;  w[j][7] = wz1 * wy1 * wx1;
    }

    // channel quads: 8 corners x ds_load_b128 (4 channels at once)
    #pragma unroll
    for (int c4 = 0; c4 < 3; ++c4) {
      v4f acc[4];
      #pragma unroll
      for (int j = 0; j < 4; ++j) {
        v4f a = {0.0f, 0.0f, 0.0f, 0.0f};
        #pragma unroll
        for (int k = 0; k < 8; ++k) {
          const v4f* vp = (const v4f*)(sgT + off[j][k] + c4 * 4);
          a += (*vp) * w[j][k];
        }
        acc[j] = a;
      }
      // 4x4 register transpose -> coalesced b128 NT stores per channel
      #pragma unroll
      for (int cc = 0; cc < 4; ++cc) {
        v4f r;
        r[0] = acc[0][cc]; r[1] = acc[1][cc];
        r[2] = acc[2][cc]; r[3] = acc[3][cc];
        __builtin_nontemporal_store(
            r, (v4f*)(gout + ((size_t)(c4 * 4 + cc) * GH + y) * GW + x4));
      }
    }
  }
}

extern "C" void kernel_launch(void* const* d_in, const int* in_sizes, int n_in,
                              void* d_out, int out_size, void* d_ws, size_t ws_size,
                              hipStream_t stream) {
  const float* grid_p  = (const float*)d_in[0];  // (16,12,8,16,16) f32
  const float* guide_p = (const float*)d_in[1];  // (16,1,512,512)  f32
  float*       out_p   = (float*)d_out;          // (16,12,512,512) f32

  const size_t need = (size_t)16 * GRID_PER_B * sizeof(float);   // 1.5 MB
  const int use_ws = (d_ws != nullptr && ws_size >= need) ? 1 : 0;

  if (use_ws) {
    // 16*24576 elements, one per thread
    transpose_kernel<<<dim3((16 * GRID_PER_B) / BLOCKT), dim3(BLOCKT), 0, stream>>>(
        grid_p, (float*)d_ws);
  }
  // 16 batches x 64 row-tiles, 256 threads (8 wave32 per block)
  slice_kernel<<<dim3(16 * 64), dim3(BLOCKT), 0, stream>>>(
      grid_p, guide_p, (const float*)d_ws, out_p, use_ws);
}